// LinearTransformer_37228776522344
// MI455X (gfx1250) — compile-verified
//
#include <hip/hip_runtime.h>
#include <hip/hip_bf16.h>

// Problem constants (from reference)
#define NLAYER 4
#define NHEAD  8
#define DP     63
#define BATCH  16
#define SEQ    2048
#define D      64

typedef __attribute__((ext_vector_type(2))) float v2f;
typedef __attribute__((ext_vector_type(8))) float v8f;

__device__ __forceinline__ v8f wmma_f32_k4(v2f a, v2f b, v8f c) {
    // D = A(16x4) * B(4x16) + C, fp32 throughout
    return __builtin_amdgcn_wmma_f32_16x16x4_f32(
        /*neg_a=*/false, a, /*neg_b=*/false, b,
        /*c_mod=*/(short)0, c, /*reuse_a=*/false, /*reuse_b=*/false);
}

// ---------------------------------------------------------------------------
// Kernel A: per-batch Gram matrix G[b] = sum_{n != SEQ-1} Z[b,n,:]^T Z[b,n,:]
// grid = BATCH*16 blocks (16 tiles of 16x16 per batch), 32 threads (1 wave).
// ---------------------------------------------------------------------------
__global__ void lt_gram_kernel(const float* __restrict__ Z, float* __restrict__ G) {
    int blk  = blockIdx.x;
    int b    = blk >> 4;
    int tile = blk & 15;
    int i0   = (tile >> 2) << 4;   // G row block
    int k0   = (tile & 3) << 4;    // G col block
    int lane = threadIdx.x;        // 0..31 (wave32)
    int half = lane >> 4;
    int l    = lane & 15;

    const float* Zb = Z + (size_t)b * SEQ * D;
    v8f c = {};
    for (int n0 = 0; n0 < SEQ; n0 += 4) {
        int r0 = n0 + 2 * half;    // even, never SEQ-1
        int r1 = r0 + 1;           // odd, can be SEQ-1 in last chunk
        // A[m=l][k] = Z[r][i0+l] (Z^T slice), B[k][n=l] = Z[r][k0+l]
        v2f a, bb;
        a.x  = Zb[(size_t)r0 * D + i0 + l];
        float ay = Zb[(size_t)r1 * D + i0 + l];
        a.y  = (r1 == SEQ - 1) ? 0.0f : ay;   // mask last token (branchless)
        bb.x = Zb[(size_t)r0 * D + k0 + l];
        bb.y = Zb[(size_t)r1 * D + k0 + l];
        c = wmma_f32_k4(a, bb, c);
    }
    float* Gb = G + (size_t)b * D * D;
#pragma unroll
    for (int v = 0; v < 8; v++) {
        int row = i0 + v + 8 * half;     // C/D layout: lanes 16-31 -> M += 8
        Gb[row * D + k0 + l] = c[v];
    }
}

// ---------------------------------------------------------------------------
// Kernel B: W[b] = I + (1/(SEQ-1)) * sum_j Q_full_j @ (G[b] @ P_full_j^T)
// params points at allparam[layer] : [head][2][63][63]
// grid = BATCH blocks, 256 threads.
// ---------------------------------------------------------------------------
__global__ void lt_mix_kernel(const float* __restrict__ G,
                              const float* __restrict__ params,
                              float* __restrict__ W) {
    __shared__ float Gs[D * D];
    __shared__ float Ps[D * D];
    __shared__ float Qs[D * D];
    __shared__ float Ts[D * D];
    int b = blockIdx.x;
    int tid = threadIdx.x;

    const float* Gb = G + (size_t)b * D * D;
    for (int e = tid; e < D * D; e += 256) Gs[e] = Gb[e];

    float acc[16];
#pragma unroll
    for (int r = 0; r < 16; r++) acc[r] = 0.0f;

    for (int j = 0; j < NHEAD; j++) {
        const float* Pp = params + (size_t)j * 2 * DP * DP;
        const float* Qp = Pp + DP * DP;
        __syncthreads();   // Gs ready (j=0); prev-iter reads of Ps/Qs/Ts done (j>0)
        // Build augmented P_full (corner=1) and Q_full (corner=0) in LDS.
        for (int e = tid; e < D * D; e += 256) {
            int r = e >> 6, cx = e & 63;
            float pv = 0.0f, qv = 0.0f;
            if (r < DP && cx < DP) { pv = Pp[r * DP + cx]; qv = Qp[r * DP + cx]; }
            if (r == DP && cx == DP) pv = 1.0f;
            Ps[e] = pv; Qs[e] = qv;
        }
        __syncthreads();
        // T = G @ P_full^T : T[i][k] = sum_l G[i][l] * P_full[k][l]
        for (int e = tid; e < D * D; e += 256) {
            int i = e >> 6, k = e & 63;
            float s = 0.0f;
            for (int l2 = 0; l2 < D; l2++) s += Gs[i * D + l2] * Ps[k * D + l2];
            Ts[e] = s;
        }
        __syncthreads();
        // acc += Q_full @ T
#pragma unroll
        for (int r = 0; r < 16; r++) {
            int e = tid + 256 * r;
            int i = e >> 6, k = e & 63;
            float s = 0.0f;
            for (int l2 = 0; l2 < D; l2++) s += Qs[i * D + l2] * Ts[l2 * D + k];
            acc[r] += s;
        }
    }
    const float inv = 1.0f / (float)(SEQ - 1);
    float* Wb = W + (size_t)b * D * D;
#pragma unroll
    for (int r = 0; r < 16; r++) {
        int e = tid + 256 * r;
        int i = e >> 6, k = e & 63;
        Wb[e] = acc[r] * inv + ((i == k) ? 1.0f : 0.0f);
    }
}

// ---------------------------------------------------------------------------
// Kernel C: Zout[b,n,:] = Zin[b,n,:] @ W[b]   (W = I + M, so this is Z + Z M)
// grid = BATCH*16 blocks, 256 threads (8 waves); each wave: 16 rows x 64 cols.
// Safe when Zin == Zout: each wave reads its 16 rows fully before storing.
// ---------------------------------------------------------------------------
__global__ void lt_apply_kernel(const float* Zin, const float* __restrict__ W,
                                float* Zout) {
    __shared__ float Ws[D * D];
    int b  = blockIdx.x >> 4;
    int rb = blockIdx.x & 15;
    int tid = threadIdx.x;

    const float* Wb = W + (size_t)b * D * D;
    for (int e = tid; e < D * D; e += 256) Ws[e] = Wb[e];
    __syncthreads();

    int wave = tid >> 5;
    int lane = tid & 31;
    int half = lane >> 4;
    int l    = lane & 15;
    int n0   = rb * 128 + wave * 16;

    const float* Zr = Zin + ((size_t)b * SEQ + n0) * D;
    v8f c0 = {}, c1 = {}, c2 = {}, c3 = {};
    for (int i0 = 0; i0 < D; i0 += 4) {
        // A[m=l][k] = Z[n0+l][i0+k] : two consecutive floats per lane
        v2f a;
        const float* ap = Zr + (size_t)l * D + i0 + 2 * half;
        a.x = ap[0];
        a.y = ap[1];
        int r0 = i0 + 2 * half;
        v2f b0, b1, b2, b3;
        b0.x = Ws[r0 * D +  0 + l]; b0.y = Ws[(r0 + 1) * D +  0 + l];
        b1.x = Ws[r0 * D + 16 + l]; b1.y = Ws[(r0 + 1) * D + 16 + l];
        b2.x = Ws[r0 * D + 32 + l]; b2.y = Ws[(r0 + 1) * D + 32 + l];
        b3.x = Ws[r0 * D + 48 + l]; b3.y = Ws[(r0 + 1) * D + 48 + l];
        c0 = wmma_f32_k4(a, b0, c0);
        c1 = wmma_f32_k4(a, b1, c1);
        c2 = wmma_f32_k4(a, b2, c2);
        c3 = wmma_f32_k4(a, b3, c3);
    }
    float* Zo = Zout + ((size_t)b * SEQ + n0) * D;
#pragma unroll
    for (int v = 0; v < 8; v++) {
        int row = v + 8 * half;
        Zo[row * D +  0 + l] = c0[v];
        Zo[row * D + 16 + l] = c1[v];
        Zo[row * D + 32 + l] = c2[v];
        Zo[row * D + 48 + l] = c3[v];
    }
}

extern "C" void kernel_launch(void* const* d_in, const int* in_sizes, int n_in,
                              void* d_out, int out_size, void* d_ws, size_t ws_size,
                              hipStream_t stream) {
    (void)in_sizes; (void)n_in; (void)out_size; (void)ws_size;
    const float* Z        = (const float*)d_in[0];   // [BATCH][SEQ][D]
    const float* allparam = (const float*)d_in[1];   // [NLAYER][NHEAD][2][DP][DP]
    float* out = (float*)d_out;                      // [BATCH][SEQ][D]

    float* G = (float*)d_ws;                 // BATCH * D * D floats (256 KB)
    float* W = G + (size_t)BATCH * D * D;    // BATCH * D * D floats (256 KB)

    const size_t layer_stride = (size_t)NHEAD * 2 * DP * DP;

    for (int layer = 0; layer < NLAYER; layer++) {
        const float* Zsrc = (layer == 0) ? Z : out;
        lt_gram_kernel<<<BATCH * 16, 32, 0, stream>>>(Zsrc, G);
        lt_mix_kernel<<<BATCH, 256, 0, stream>>>(G, allparam + layer * layer_stride, W);
        lt_apply_kernel<<<BATCH * 16, 256, 0, stream>>>(Zsrc, W, out);
    }
}